// ComplexQuantumLayer_26328149525142
// MI455X (gfx1250) — compile-verified
//
#include <hip/hip_runtime.h>
#include <hip/hip_bf16.h>
#include <math.h>

// Problem constants (from reference)
#define NQ    10
#define NL    6
#define BATCH 2048
#define DIM   1024   // 2^NQ

// Workspace layout (d_ws), all float32:
//   psi_re : BATCH*DIM   (8 MB)
//   psi_im : BATCH*DIM   (8 MB)
//   M_re   : DIM*DIM     (4 MB)   M[j,k] = U_total[k,j]  (so Y = psi * M)
//   M_im   : DIM*DIM     (4 MB)
// total 24 MB scratch.

typedef float v2f __attribute__((ext_vector_type(2)));
typedef float v8f __attribute__((ext_vector_type(8)));

__device__ __forceinline__ float2 cmul(float2 a, float2 b) {
    return make_float2(a.x * b.x - a.y * b.y, a.x * b.y + a.y * b.x);
}
__device__ __forceinline__ float2 cadd(float2 a, float2 b) {
    return make_float2(a.x + b.x, a.y + b.y);
}
__device__ __forceinline__ float2 cscale(float s, float2 a) {
    return make_float2(s * a.x, s * a.y);
}

// ---------------------------------------------------------------------------
// Kernel 1: AngleEmbedding product state.
// psi[b,k] = prod_q (bit_q(k) ? sin(x/2) : cos(x/2)) * (-i)^popcount(k)
// wire q <-> bit (NQ-1-q) (PennyLane: wire 0 is MSB).
// ---------------------------------------------------------------------------
__global__ void __launch_bounds__(256) embed_kernel(const float* __restrict__ x,
                                                    float* __restrict__ pre,
                                                    float* __restrict__ pim) {
    __shared__ float cs[NQ], sn[NQ];
    const int b = blockIdx.x;
    const int tid = threadIdx.x;
    if (tid < NQ) {
        float s, c;
        sincosf(0.5f * x[b * NQ + tid], &s, &c);
        sn[tid] = s; cs[tid] = c;
    }
    __syncthreads();
    #pragma unroll
    for (int i = 0; i < DIM / 256; ++i) {
        const int k = tid + 256 * i;
        float r = 1.0f;
        #pragma unroll
        for (int q = 0; q < NQ; ++q) {
            const int bit = (k >> (NQ - 1 - q)) & 1;
            r *= bit ? sn[q] : cs[q];
        }
        const int p = __popc(k) & 3;            // phase (-i)^p
        const float re = (p == 0) ? r : (p == 2 ? -r : 0.0f);
        const float im = (p == 1) ? -r : (p == 3 ? r : 0.0f);
        pre[b * DIM + k] = re;
        pim[b * DIM + k] = im;
    }
}

// ---------------------------------------------------------------------------
// Kernel 2: build M[j,:] = (U_total e_j)^T by simulating basis state |j> in LDS.
// One block per basis state; state = 1024 complex in 8 KB LDS.
// ---------------------------------------------------------------------------
__global__ void __launch_bounds__(128) build_u_kernel(const float* __restrict__ w,
                                                      float* __restrict__ Mre,
                                                      float* __restrict__ Mim) {
    __shared__ float2 st[DIM];
    const int j = blockIdx.x;
    const int tid = threadIdx.x;
    for (int k = tid; k < DIM; k += 128)
        st[k] = make_float2(k == j ? 1.0f : 0.0f, 0.0f);
    __syncthreads();

    for (int l = 0; l < NL; ++l) {
        // ---- Rot(phi,theta,omega) on each wire ----
        for (int q = 0; q < NQ; ++q) {
            const float phi = w[(l * NQ + q) * 3 + 0];
            const float th  = w[(l * NQ + q) * 3 + 1];
            const float om  = w[(l * NQ + q) * 3 + 2];
            float ct, s_t, ca, sa, cb, sb;
            sincosf(0.5f * th, &s_t, &ct);
            sincosf(0.5f * (phi + om), &sa, &ca);
            sincosf(0.5f * (phi - om), &sb, &cb);
            const float2 ga = make_float2( ct * ca, -ct * sa);   // e^{-i(phi+om)/2} cos
            const float2 gb = make_float2(-s_t * cb, -s_t * sb); // -e^{+i(phi-om)/2} sin
            const float2 gc = make_float2( s_t * cb, -s_t * sb); //  e^{-i(phi-om)/2} sin
            const float2 gd = make_float2( ct * ca,  ct * sa);   // e^{+i(phi+om)/2} cos
            const int p = NQ - 1 - q;
            for (int t = tid; t < DIM / 2; t += 128) {
                const int i0 = ((t >> p) << (p + 1)) | (t & ((1 << p) - 1));
                const int i1 = i0 | (1 << p);
                const float2 s0 = st[i0], s1 = st[i1];
                st[i0] = cadd(cmul(ga, s0), cmul(gb, s1));
                st[i1] = cadd(cmul(gc, s0), cmul(gd, s1));
            }
            __syncthreads();
        }
        // ---- CNOT ring: control q, target (q+1)%NQ (pure amplitude swap) ----
        for (int q = 0; q < NQ; ++q) {
            const int pc = NQ - 1 - q;
            const int pt = NQ - 1 - ((q + 1) % NQ);
            const int plo = pc < pt ? pc : pt;
            const int phi_ = pc < pt ? pt : pc;
            for (int t = tid; t < DIM / 4; t += 128) {
                int i = ((t >> plo) << (plo + 1)) | (t & ((1 << plo) - 1));
                i = ((i >> phi_) << (phi_ + 1)) | (i & ((1 << phi_) - 1));
                const int i1 = i | (1 << pc);        // control=1, target=0
                const int i2 = i1 | (1 << pt);       // control=1, target=1
                const float2 tmp = st[i1]; st[i1] = st[i2]; st[i2] = tmp;
            }
            __syncthreads();
        }
        // ---- IsingXX(w[l,0,0]) on wires (0,1) -> bits 9,8, mask 0x300 ----
        {
            float s, c; sincosf(0.5f * w[(l * NQ + 0) * 3 + 0], &s, &c);
            const float2 ms = make_float2(0.0f, -s);           // -i s
            for (int t = tid; t < DIM / 2; t += 128) {         // t has bit9==0
                const int jd = t ^ 0x300;
                const float2 a0 = st[t], a1 = st[jd];
                st[t]  = cadd(cscale(c, a0), cmul(ms, a1));
                st[jd] = cadd(cscale(c, a1), cmul(ms, a0));
            }
            __syncthreads();
        }
        // ---- IsingYY(w[l,0,1]) on wires (1,2) -> bits 8,7, mask 0x180 ----
        {
            float s, c; sincosf(0.5f * w[(l * NQ + 0) * 3 + 1], &s, &c);
            for (int t = tid; t < DIM / 2; t += 128) {
                const int i = ((t >> 8) << 9) | (t & 0xFF);    // bit8 == 0
                const int jd = i ^ 0x180;
                // sigma = -1 if the two bits equal (b7==0 for i), else +1
                const float sgn = ((i >> 7) & 1) ? 1.0f : -1.0f;
                const float2 ms = make_float2(0.0f, -sgn * s); // sigma * (-i s)
                const float2 a0 = st[i], a1 = st[jd];
                st[i]  = cadd(cscale(c, a0), cmul(ms, a1));
                st[jd] = cadd(cscale(c, a1), cmul(ms, a0));
            }
            __syncthreads();
        }
        // ---- IsingZZ(w[l,0,2]) on wires (2,3) -> bits 7,6 (diagonal) ----
        {
            float s, c; sincosf(0.5f * w[(l * NQ + 0) * 3 + 2], &s, &c);
            for (int k = tid; k < DIM; k += 128) {
                const int df = ((k >> 7) ^ (k >> 6)) & 1;       // bits differ?
                const float2 ph = make_float2(c, df ? s : -s);  // e^{±i th/2}
                st[k] = cmul(ph, st[k]);
            }
            __syncthreads();
        }
    }

    for (int k = tid; k < DIM; k += 128) {
        Mre[j * DIM + k] = st[k].x;
        Mim[j * DIM + k] = st[k].y;
    }
}

// ---------------------------------------------------------------------------
// Kernel 3: zero the accumulated output (harness poisons d_out).
// ---------------------------------------------------------------------------
__global__ void zero_kernel(float* __restrict__ out, int n) {
    const int i = blockIdx.x * blockDim.x + threadIdx.x;
    if (i < n) out[i] = 0.0f;
}

// ---------------------------------------------------------------------------
// Kernel 4: complex GEMM Y = psi * M via V_WMMA_F32_16X16X4_F32, fused with
// |Y|^2 and signed PauliZ reductions into out[b,q].
// One wave32 per 16x16 output tile; 8 waves (256 threads) per block.
// Complex MAC per K-step of 4:
//   acc_re += Ar*Br + (-Ai)*Bi     (A_im negated in VALU: float-WMMA NEG is C-neg only)
//   acc_im += Ar*Bi +   Ai *Br
// Fragment layouts per CDNA5 ISA:
//   A 16x4 : lane%16 = M, K = 2*(lane/16)+j  (j = VGPR idx 0/1) -> float2 load
//   B 4x16 : lane%16 = N, K = 2*(lane/16)+j  -> two strided b32 loads
//   C 16x16: VGPR v, lane l -> M = v + 8*(l/16), N = l%16
// ---------------------------------------------------------------------------
__global__ void __launch_bounds__(256) gemm_z_kernel(const float* __restrict__ Are,
                                                     const float* __restrict__ Aim,
                                                     const float* __restrict__ Bre,
                                                     const float* __restrict__ Bim,
                                                     float* __restrict__ out) {
    const int lane = threadIdx.x & 31;
    const int wave = threadIdx.x >> 5;
    const int gw = blockIdx.x * 8 + wave;        // global wave id
    const int tile_b = gw >> 6;                  // BATCH/16 = 128 tiles
    const int tile_u = gw & 63;                  // DIM/16   =  64 tiles
    const int b0 = tile_b * 16;
    const int u0 = tile_u * 16;
    const int half = lane >> 4;                  // 0: lanes 0-15, 1: lanes 16-31
    const int l16 = lane & 15;
    const int ucol = u0 + l16;

    const float* arp = Are + (b0 + l16) * DIM + 2 * half;   // A row, k-contig
    const float* aip = Aim + (b0 + l16) * DIM + 2 * half;
    const float* brp = Bre + (2 * half) * DIM + ucol;       // B col, k-strided
    const float* bip = Bim + (2 * half) * DIM + ucol;

    v8f acc_re = {0.f, 0.f, 0.f, 0.f, 0.f, 0.f, 0.f, 0.f};
    v8f acc_im = {0.f, 0.f, 0.f, 0.f, 0.f, 0.f, 0.f, 0.f};

    #pragma unroll 4
    for (int k0 = 0; k0 < DIM; k0 += 4) {
        v2f ar = *(const v2f*)(arp + k0);
        v2f ai = *(const v2f*)(aip + k0);
        v2f br = { brp[k0 * DIM], brp[(k0 + 1) * DIM] };
        v2f bi = { bip[k0 * DIM], bip[(k0 + 1) * DIM] };
        v2f nai = { -ai[0], -ai[1] };

        acc_re = __builtin_amdgcn_wmma_f32_16x16x4_f32(false, ar,  false, br, (short)0, acc_re, false, false);
        acc_re = __builtin_amdgcn_wmma_f32_16x16x4_f32(false, nai, false, bi, (short)0, acc_re, false, false);
        acc_im = __builtin_amdgcn_wmma_f32_16x16x4_f32(false, ar,  false, bi, (short)0, acc_im, false, false);
        acc_im = __builtin_amdgcn_wmma_f32_16x16x4_f32(false, ai,  false, br, (short)0, acc_im, false, false);

        // Speculative prefetch of the next K chunk (gfx1250 global_prefetch).
        __builtin_prefetch(arp + k0 + 64, 0, 1);
        __builtin_prefetch(brp + (k0 + 16) * DIM, 0, 1);
    }

    // Epilogue: probabilities, then signed column reduction per qubit.
    float p[8];
    #pragma unroll
    for (int v = 0; v < 8; ++v)
        p[v] = acc_re[v] * acc_re[v] + acc_im[v] * acc_im[v];

    #pragma unroll
    for (int q = 0; q < NQ; ++q) {
        const float sgn = ((ucol >> (NQ - 1 - q)) & 1) ? -1.0f : 1.0f;
        float t[8];
        #pragma unroll
        for (int v = 0; v < 8; ++v) t[v] = sgn * p[v];
        // reduce over the 16 columns held by each 16-lane half (masks < 16
        // keep the shuffle inside the half -> rows 0-7 / 8-15 stay separate)
        #pragma unroll
        for (int m = 1; m < 16; m <<= 1) {
            #pragma unroll
            for (int v = 0; v < 8; ++v) t[v] += __shfl_xor(t[v], m, 32);
        }
        if (l16 == 0) {
            #pragma unroll
            for (int v = 0; v < 8; ++v)
                atomicAdd(&out[(b0 + half * 8 + v) * NQ + q], t[v]);
        }
    }
}

// ---------------------------------------------------------------------------
extern "C" void kernel_launch(void* const* d_in, const int* in_sizes, int n_in,
                              void* d_out, int out_size, void* d_ws, size_t ws_size,
                              hipStream_t stream) {
    const float* x = (const float*)d_in[0];      // (2048, 10) f32
    const float* w = (const float*)d_in[1];      // (6, 10, 3) f32
    float* out = (float*)d_out;                  // (2048, 10) f32

    float* pre = (float*)d_ws;                   // needs 24 MB of d_ws
    float* pim = pre + (size_t)BATCH * DIM;
    float* Mre = pim + (size_t)BATCH * DIM;
    float* Mim = Mre + (size_t)DIM * DIM;

    embed_kernel  <<<BATCH, 256, 0, stream>>>(x, pre, pim);
    build_u_kernel<<<DIM,   128, 0, stream>>>(w, Mre, Mim);
    zero_kernel   <<<(BATCH * NQ + 255) / 256, 256, 0, stream>>>(out, BATCH * NQ);
    gemm_z_kernel <<<(BATCH / 16) * (DIM / 16) / 8, 256, 0, stream>>>(pre, pim, Mre, Mim, out);
}